// Sub_Clust_Cos_GLM_44495861186824
// MI455X (gfx1250) — compile-verified
//
#include <hip/hip_runtime.h>
#include <hip/hip_bf16.h>
#include <math.h>

// ---------------- problem constants ----------------
#define SUB_NO   20
#define E_NO     2000
#define I_NO     500
#define T_NO     200
#define HID_NO   10
#define BATCH    8
#define T_DATA   20000
#define NBASIS   24

// output layout (floats): final | C_syn_e | C_syn_i | log_C_syn_e | log_C_syn_i
#define OUT_FINAL 0
#define OUT_CSE   160000
#define OUT_CSI   200000
#define OUT_LCSE  210000
#define OUT_LCSI  250000

// workspace layout (floats)
#define WS_CFOLD  0                         // [32][E_NO]  (rows 20..31 zero)
#define WS_KERNA  64000                     // [SUB][16][T_NO] (h rows 10..15 zero)
#define WS_EXPW2  128000                    // [SUB][16]   (h 10..15 zero)
#define WS_B1P    128320                    // [SUB][16]   (h 10..15 zero)
#define WS_SYN    128640                    // [BATCH][SUB][ROWLEN] causal-padded syn_e
#define ROWLEN    20208                     // 199 head pad + 20000 + 9 tail pad

#define KC        200                       // K-chunk staged in LDS (divides E_NO)
#define NTILES    (T_DATA / 16)             // 1250 16-t tiles per batch

typedef float v2f __attribute__((ext_vector_type(2)));
typedef float v8f __attribute__((ext_vector_type(8)));

static __device__ __forceinline__ v8f wmma4(v2f a, v2f b, v8f c) {
    // V_WMMA_F32_16X16X4_F32 : D = A(16x4) * B(4x16) + C
    return __builtin_amdgcn_wmma_f32_16x16x4_f32(
        /*neg_a=*/false, a, /*neg_b=*/false, b,
        /*c_mod=*/(short)0, c, /*reuse_a=*/false, /*reuse_b=*/false);
}

// ---------------- kernel 1: softmax / log-softmax + folded cluster matrix ----
__global__ void k_softmax(const float* __restrict__ Ce_raw,
                          const float* __restrict__ Ci_raw,
                          const float* __restrict__ tempP,
                          const float* __restrict__ E_scale,
                          float* __restrict__ out,
                          float* __restrict__ ws) {
    int id = blockIdx.x * blockDim.x + threadIdx.x;
    if (id >= E_NO + I_NO) return;
    float invt = 1.0f / tempP[0];
    if (id < E_NO) {
        int col = id;
        float r[SUB_NO];
        float mx = -INFINITY;
        #pragma unroll
        for (int s = 0; s < SUB_NO; ++s) {
            r[s] = Ce_raw[s * E_NO + col] * invt;
            mx = fmaxf(mx, r[s]);
        }
        float sum = 0.0f;
        #pragma unroll
        for (int s = 0; s < SUB_NO; ++s) sum += __expf(r[s] - mx);
        float lse = __logf(sum) + mx;
        float es = __expf(E_scale[col]);
        #pragma unroll
        for (int s = 0; s < SUB_NO; ++s) {
            float lg = r[s] - lse;
            float p  = __expf(lg);
            out[OUT_CSE  + s * E_NO + col] = p;
            out[OUT_LCSE + s * E_NO + col] = lg;
            ws[WS_CFOLD + s * E_NO + col]  = p * es;
        }
        #pragma unroll
        for (int s = SUB_NO; s < 32; ++s)       // zero pad rows for 32-wide WMMA M tile
            ws[WS_CFOLD + s * E_NO + col] = 0.0f;
    } else {
        int col = id - E_NO;
        float r[SUB_NO];
        float mx = -INFINITY;
        #pragma unroll
        for (int s = 0; s < SUB_NO; ++s) {
            r[s] = Ci_raw[s * I_NO + col] * invt;
            mx = fmaxf(mx, r[s]);
        }
        float sum = 0.0f;
        #pragma unroll
        for (int s = 0; s < SUB_NO; ++s) sum += __expf(r[s] - mx);
        float lse = __logf(sum) + mx;
        #pragma unroll
        for (int s = 0; s < SUB_NO; ++s) {
            float lg = r[s] - lse;
            out[OUT_CSI  + s * I_NO + col] = __expf(lg);
            out[OUT_LCSI + s * I_NO + col] = lg;
        }
    }
}

// ---------------- kernel 2: conv kernels from basis + padded exp(W2)/bias ----
__global__ void k_prep(const float* __restrict__ W1,        // [SUB*HID][NBASIS]
                       const float* __restrict__ basis,     // [NBASIS][T_NO]
                       const float* __restrict__ W2,        // [SUB][HID]
                       const float* __restrict__ b1,        // [SUB*HID]
                       float* __restrict__ ws) {
    int id = blockIdx.x * blockDim.x + threadIdx.x;
    if (id < SUB_NO * 16 * T_NO) {                          // kernA[s][h][k]
        int s   = id / (16 * T_NO);
        int rem = id % (16 * T_NO);
        int h   = rem / T_NO;
        int k   = rem % T_NO;
        float v = 0.0f;
        if (h < HID_NO) {
            const float* wrow = W1 + (s * HID_NO + h) * NBASIS;
            #pragma unroll
            for (int i = 0; i < NBASIS; ++i) v += wrow[i] * basis[i * T_NO + k];
        }
        ws[WS_KERNA + id] = v;
    } else if (id < SUB_NO * 16 * T_NO + SUB_NO * 16) {
        int j = id - SUB_NO * 16 * T_NO;
        int s = j / 16, h = j % 16;
        ws[WS_EXPW2 + j] = (h < HID_NO) ? __expf(W2[s * HID_NO + h]) : 0.0f;
    } else if (id < SUB_NO * 16 * T_NO + 2 * SUB_NO * 16) {
        int j = id - SUB_NO * 16 * T_NO - SUB_NO * 16;
        int s = j / 16, h = j % 16;
        ws[WS_B1P + j] = (h < HID_NO) ? b1[s * HID_NO + h] : 0.0f;
    }
}

// ---------------- kernel 3: zero causal head / tail pads of syn rows --------
__global__ void k_zeropad(float* __restrict__ ws) {
    int id = blockIdx.x * blockDim.x + threadIdx.x;
    const int PER_ROW = 208;                                // 199 head + 9 tail
    if (id >= BATCH * SUB_NO * PER_ROW) return;
    int row = id / PER_ROW;
    int j   = id % PER_ROW;
    int idx = (j < T_NO - 1) ? j : (T_DATA + j);            // [0,199) or [20199,20208)
    ws[WS_SYN + (size_t)row * ROWLEN + idx] = 0.0f;
}

// ---------------- kernel 4: clustering GEMM  syn_e[b,s,t] (f32 WMMA) --------
// A (Cfold, 32 subunit rows) staged per-workgroup in LDS in K-chunks.
// Each wave owns TWO 16-t tiles: per K-step 2 ds_load + 2 global_load + 4 WMMA.
__global__ void __launch_bounds__(256) k_syn_gemm(const float* __restrict__ S_e,
                                                  float* __restrict__ ws) {
    __shared__ float lA[32 * KC];                           // 25.6 KB
    const float* Cf = ws + WS_CFOLD;
    float* syn = ws + WS_SYN;
    int wave = threadIdx.x >> 5;
    int lane = threadIdx.x & 31;
    int l15  = lane & 15;
    int kb   = (lane >> 4) * 2;                             // K sub-offset per half-wave
    int b    = blockIdx.y;
    int tile0 = blockIdx.x * 16 + wave * 2;                 // two adjacent tiles per wave
    int tile1 = tile0 + 1;
    bool valid0 = (tile0 < NTILES);
    bool valid1 = (tile1 < NTILES);
    int tc0 = valid0 ? tile0 : 0;                           // clamp for safe loads
    int tc1 = valid1 ? tile1 : 0;
    const float* brow0 = S_e + ((size_t)b * T_DATA + tc0 * 16 + l15) * E_NO;
    const float* brow1 = S_e + ((size_t)b * T_DATA + tc1 * 16 + l15) * E_NO;
    v8f c00 = {}, c01 = {}, c10 = {}, c11 = {};
    for (int k0 = 0; k0 < E_NO; k0 += KC) {
        __syncthreads();                                    // previous chunk consumed
        for (int i = threadIdx.x; i < 32 * KC; i += 256) {  // coop stage A chunk
            int row = i / KC, col = i % KC;
            lA[i] = Cf[(size_t)row * E_NO + k0 + col];
        }
        __syncthreads();
        if (k0 + KC < E_NO) {                               // near-cache stream prefetch
            __builtin_prefetch(brow0 + k0 + KC, 0, 3);
            __builtin_prefetch(brow1 + k0 + KC, 0, 3);
        }
        const float* pb0 = brow0 + k0 + kb;
        const float* pb1 = brow1 + k0 + kb;
        #pragma unroll 10
        for (int k = 0; k < KC; k += 4) {
            v2f a0 = *(const v2f*)&lA[l15 * KC + kb + k];         // s = 0..15
            v2f a1 = *(const v2f*)&lA[(16 + l15) * KC + kb + k];  // s = 16..31 (20+ zero)
            v2f b0 = *(const v2f*)(pb0 + k);
            v2f b1 = *(const v2f*)(pb1 + k);
            c00 = wmma4(a0, b0, c00);
            c01 = wmma4(a1, b0, c01);
            c10 = wmma4(a0, b1, c10);
            c11 = wmma4(a1, b1, c11);
        }
    }
    // scatter D tiles into causally padded syn rows
    float* base = syn + (size_t)b * SUB_NO * ROWLEN;
    int moff = (lane < 16) ? 0 : 8;
    if (valid0) {
        int t = tile0 * 16 + l15;
        #pragma unroll
        for (int r = 0; r < 8; ++r)
            base[(size_t)(r + moff) * ROWLEN + (T_NO - 1) + t] = c00[r];
        if (lane < 16) {
            #pragma unroll
            for (int r = 0; r < 4; ++r)                     // s = 16..19 only
                base[(size_t)(16 + r) * ROWLEN + (T_NO - 1) + t] = c01[r];
        }
    }
    if (valid1) {
        int t = tile1 * 16 + l15;
        #pragma unroll
        for (int r = 0; r < 8; ++r)
            base[(size_t)(r + moff) * ROWLEN + (T_NO - 1) + t] = c10[r];
        if (lane < 16) {
            #pragma unroll
            for (int r = 0; r < 4; ++r)
                base[(size_t)(16 + r) * ROWLEN + (T_NO - 1) + t] = c11[r];
        }
    }
}

// ---------------- kernel 5: causal conv (Toeplitz WMMA) + tanh + mix + sum --
// workgroup: one (batch, 128-t chunk); wave w owns t-tile chunk+16w; loops subunits
__global__ void k_conv_out(float* __restrict__ out,
                           const float* __restrict__ ws,
                           const float* __restrict__ VoP) {
    __shared__ float win[336];                              // syn window: 199 + 128 = 327
    int b     = blockIdx.y;
    int cbase = blockIdx.x * 128;
    int wave  = threadIdx.x >> 5;
    int lane  = threadIdx.x & 31;
    int l15   = lane & 15;
    int kb    = (lane >> 4) * 2;
    int t0    = cbase + wave * 16;
    bool active = (t0 < T_DATA);
    const float* kernA = ws + WS_KERNA;
    const float* syn   = ws + WS_SYN;
    const float* eW    = ws + WS_EXPW2;
    const float* b1p   = ws + WS_B1P;
    float acc = 0.0f;
    for (int s = 0; s < SUB_NO; ++s) {
        const float* row = syn + (size_t)(b * SUB_NO + s) * ROWLEN;
        __syncthreads();                                    // protect previous window
        for (int j = threadIdx.x; j < 327; j += 256) {
            int g = cbase + j;
            win[j] = (g < ROWLEN) ? row[g] : 0.0f;
        }
        __syncthreads();
        if (active) {
            v8f c = {};
            const float* arow = kernA + (size_t)(s * 16 + l15) * T_NO;
            int tbase = (t0 - cbase) + l15 + (T_NO - 1);    // LDS index of syn_e(t) at k=0
            #pragma unroll 10
            for (int k0 = 0; k0 < T_NO; k0 += 4) {
                v2f a = *(const v2f*)(arow + k0 + kb);
                int idx = tbase - k0 - kb;
                v2f bb;
                bb[0] = win[idx];                           // syn_e(t - (k0+kb))
                bb[1] = win[idx - 1];                       // syn_e(t - (k0+kb) - 1)
                c = wmma4(a, bb, c);
            }
            int moff = (lane < 16) ? 0 : 8;
            #pragma unroll
            for (int r = 0; r < 8; ++r) {
                int m = r + moff;                           // hid row (>=10 rows exact 0)
                acc += eW[s * 16 + m] * tanhf(c[r] + b1p[s * 16 + m]);
            }
        }
    }
    if (active) {
        float other = __shfl_xor(acc, 16, 32);              // fold the two 16-lane halves
        if (lane < 16)
            out[OUT_FINAL + (size_t)b * T_DATA + t0 + l15] = acc + other + VoP[0];
    }
}

// ---------------- launch -----------------------------------------------------
extern "C" void kernel_launch(void* const* d_in, const int* in_sizes, int n_in,
                              void* d_out, int out_size, void* d_ws, size_t ws_size,
                              hipStream_t stream) {
    const float* S_e      = (const float*)d_in[0];
    // d_in[1] = S_i   (dead code w.r.t. outputs)
    const float* temp     = (const float*)d_in[2];
    const float* E_scale  = (const float*)d_in[3];
    // d_in[4] = I_scale (dead code)
    const float* W1       = (const float*)d_in[5];
    const float* W2       = (const float*)d_in[6];
    const float* b1       = (const float*)d_in[7];
    const float* Ce_raw   = (const float*)d_in[8];
    const float* Ci_raw   = (const float*)d_in[9];
    const float* V_o      = (const float*)d_in[10];
    const float* basis    = (const float*)d_in[11];
    float* out = (float*)d_out;
    float* ws  = (float*)d_ws;

    k_softmax<<<(E_NO + I_NO + 255) / 256, 256, 0, stream>>>(Ce_raw, Ci_raw, temp, E_scale, out, ws);
    k_prep<<<(SUB_NO * 16 * T_NO + 2 * SUB_NO * 16 + 255) / 256, 256, 0, stream>>>(W1, basis, W2, b1, ws);
    k_zeropad<<<(BATCH * SUB_NO * 208 + 255) / 256, 256, 0, stream>>>(ws);

    int wg_per_b_gemm = (NTILES + 15) / 16;                  // 79 (16 tiles per WG)
    k_syn_gemm<<<dim3(wg_per_b_gemm, BATCH), 256, 0, stream>>>(S_e, ws);

    int wg_per_b_conv = (T_DATA + 127) / 128;                // 157
    k_conv_out<<<dim3(wg_per_b_conv, BATCH), 256, 0, stream>>>(out, ws, V_o);
}